// MultiScalePatchMambaClassifier_15083925143613
// MI455X (gfx1250) — compile-verified
//
#include <hip/hip_runtime.h>

typedef __attribute__((ext_vector_type(2))) float v2f;
typedef __attribute__((ext_vector_type(8))) float v8f;

#define B_SZ    4
#define L_SEQ   2048
#define D_MODEL 512
#define D_INNER 1024
#define D_STATE 16
#define DT_RANK 32

__device__ __forceinline__ v8f wmma_f32_k4(v2f a, v2f b, v8f c) {
  // V_WMMA_F32_16X16X4_F32 : D(16x16,f32) = A(16x4,f32) * B(4x16,f32) + C
  return __builtin_amdgcn_wmma_f32_16x16x4_f32(false, a, false, b, (short)0, c,
                                               false, false);
}

// scan-order permutations: 0=fwd, 1=rev, 2=even-first, 3=odd-first
__device__ __forceinline__ int perm_idx(int k, int j) {
  switch (k) {
    case 0:  return j;
    case 1:  return L_SEQ - 1 - j;
    case 2:  return (j < L_SEQ / 2) ? 2 * j : 2 * (j - L_SEQ / 2) + 1;
    default: return (j < L_SEQ / 2) ? 2 * j + 1 : 2 * (j - L_SEQ / 2);
  }
}

// ---------------- K1: in_proj GEMM (8192 x 2048 x 512), split + silu(z) ----
// 2x2 register blocking: each wave computes a 32x32 output block.
__global__ __launch_bounds__(128) void k_in_proj(
    const float* __restrict__ x, const float* __restrict__ w,
    float* __restrict__ x_ssm, float* __restrict__ z_silu) {
  int wave = threadIdx.x >> 5, lane = threadIdx.x & 31;
  int tile = blockIdx.x * 4 + wave;        // 16384 waves total
  int nt = tile & 63;                      // 2048/32 col super-tiles
  int mt = tile >> 6;                      // 8192/32 row super-tiles
  int m0 = mt * 32, n0 = nt * 32;
  int lmod = lane & 15, lhalf = lane >> 4;
  const float* a0 = x + (size_t)(m0 + lmod) * D_MODEL + 2 * lhalf;
  const float* a1 = x + (size_t)(m0 + 16 + lmod) * D_MODEL + 2 * lhalf;
  const float* b0 = w + (size_t)(n0 + lmod) * D_MODEL + 2 * lhalf;
  const float* b1 = w + (size_t)(n0 + 16 + lmod) * D_MODEL + 2 * lhalf;
  v8f acc00 = {0.f,0.f,0.f,0.f,0.f,0.f,0.f,0.f};
  v8f acc01 = acc00, acc10 = acc00, acc11 = acc00;
#pragma unroll 2
  for (int kk = 0; kk < D_MODEL; kk += 4) {
    v2f va0 = *(const v2f*)(a0 + kk);
    v2f va1 = *(const v2f*)(a1 + kk);
    v2f vb0 = *(const v2f*)(b0 + kk);
    v2f vb1 = *(const v2f*)(b1 + kk);
    acc00 = wmma_f32_k4(va0, vb0, acc00);
    acc01 = wmma_f32_k4(va0, vb1, acc01);
    acc10 = wmma_f32_k4(va1, vb0, acc10);
    acc11 = wmma_f32_k4(va1, vb1, acc11);
  }
  v8f accs[4] = {acc00, acc01, acc10, acc11};
#pragma unroll
  for (int s = 0; s < 4; ++s) {
    int mb = s >> 1, nb = s & 1;
    int col = n0 + nb * 16 + lmod;   // D_INNER % 32 == 0: no straddle
#pragma unroll
    for (int r = 0; r < 8; ++r) {
      int row = m0 + mb * 16 + 8 * lhalf + r;
      float v = accs[s][r];
      if (col < D_INNER) {
        x_ssm[(size_t)row * D_INNER + col] = v;
      } else {
        z_silu[(size_t)row * D_INNER + (col - D_INNER)] =
            v / (1.f + __expf(-v));
      }
    }
  }
}

// ---------------- K2: depthwise conv (k=4, pad 1/2) + silu --------------
__global__ __launch_bounds__(256) void k_conv(
    const float* __restrict__ x_ssm, const float* __restrict__ cw,
    const float* __restrict__ cb, float* __restrict__ xcl) {
  size_t idx = (size_t)blockIdx.x * 256 + threadIdx.x;
  int d = (int)(idx & (D_INNER - 1));
  size_t t = idx >> 10;                 // token = b*L + l
  int l = (int)(t & (L_SEQ - 1));
  size_t tb = t - (size_t)l;            // b*L
  float acc = cb[d];
#pragma unroll
  for (int j = 0; j < 4; ++j) {
    int ls = l - 1 + j;
    if (ls >= 0 && ls < L_SEQ)
      acc += cw[d * 4 + j] * x_ssm[(tb + ls) * D_INNER + d];
  }
  xcl[idx] = acc / (1.f + __expf(-acc));
}

// ---------------- K3: x_proj GEMM per direction (rows permuted) ----------
// x_dbl[k,b,j,c] = sum_d xcl[b, perm_k(j), d] * Wx[k,c,d]   (c: 0..63)
__global__ __launch_bounds__(128) void k_x_proj(
    const float* __restrict__ xcl, const float* __restrict__ wx,
    float* __restrict__ x_dbl) {
  int wave = threadIdx.x >> 5, lane = threadIdx.x & 31;
  int tile = blockIdx.x * 4 + wave;   // 2048 waves total
  int ct = tile & 1;  tile >>= 1;     // 64/32 col super-tiles
  int jt = tile & 63; tile >>= 6;     // 2048/32 row super-tiles
  int b = tile & 3;   int k = tile >> 2;
  int j0 = jt * 32, c0 = ct * 32;
  int lmod = lane & 15, lhalf = lane >> 4;
  int p0 = perm_idx(k, j0 + lmod);
  int p1 = perm_idx(k, j0 + 16 + lmod);
  const float* a0 = xcl + ((size_t)b * L_SEQ + p0) * D_INNER + 2 * lhalf;
  const float* a1 = xcl + ((size_t)b * L_SEQ + p1) * D_INNER + 2 * lhalf;
  const float* b0 = wx + ((size_t)k * 64 + c0 + lmod) * D_INNER + 2 * lhalf;
  const float* b1 = wx + ((size_t)k * 64 + c0 + 16 + lmod) * D_INNER + 2 * lhalf;
  v8f acc00 = {0.f,0.f,0.f,0.f,0.f,0.f,0.f,0.f};
  v8f acc01 = acc00, acc10 = acc00, acc11 = acc00;
#pragma unroll 2
  for (int kk = 0; kk < D_INNER; kk += 4) {
    v2f va0 = *(const v2f*)(a0 + kk);
    v2f va1 = *(const v2f*)(a1 + kk);
    v2f vb0 = *(const v2f*)(b0 + kk);
    v2f vb1 = *(const v2f*)(b1 + kk);
    acc00 = wmma_f32_k4(va0, vb0, acc00);
    acc01 = wmma_f32_k4(va0, vb1, acc01);
    acc10 = wmma_f32_k4(va1, vb0, acc10);
    acc11 = wmma_f32_k4(va1, vb1, acc11);
  }
  size_t base = (size_t)(k * B_SZ + b) * L_SEQ;
  v8f accs[4] = {acc00, acc01, acc10, acc11};
#pragma unroll
  for (int s = 0; s < 4; ++s) {
    int mb = s >> 1, nb = s & 1;
#pragma unroll
    for (int r = 0; r < 8; ++r) {
      int jr = j0 + mb * 16 + 8 * lhalf + r;
      x_dbl[(base + jr) * 64 + c0 + nb * 16 + lmod] = accs[s][r];
    }
  }
}

// ---------------- K5: selective scan (fused dt_proj + softplus) ----------
// one lane per (k,b,d) channel; 16-state recurrence; writes per-direction
// output already at the MERGED position p = perm_k(j)
__global__ __launch_bounds__(256) void k_scan(
    const float* __restrict__ xcl, const float* __restrict__ x_dbl,
    const float* __restrict__ dtw, const float* __restrict__ dtb,
    const float* __restrict__ a_logs, const float* __restrict__ dsv,
    float* __restrict__ ypart) {
  int bid = blockIdx.x;
  int dblk = bid & 3; bid >>= 2;
  int k = bid & 3;  int b = bid >> 2;
  int d = dblk * 256 + threadIdx.x;
  int kd = k * D_INNER + d;
  float A[D_STATE], h[D_STATE], W[DT_RANK];
#pragma unroll
  for (int n = 0; n < D_STATE; ++n) {
    A[n] = -__expf(a_logs[(size_t)kd * D_STATE + n]);
    h[n] = 0.f;
  }
#pragma unroll
  for (int r = 0; r < DT_RANK; ++r) W[r] = dtw[(size_t)kd * DT_RANK + r];
  float bias = dtb[kd];
  float Dd = dsv[kd];
  const float* xd = x_dbl + (size_t)(k * B_SZ + b) * L_SEQ * 64;
  const float* xu = xcl + (size_t)b * L_SEQ * D_INNER;
  float* yo = ypart + (size_t)(k * B_SZ + b) * L_SEQ * D_INNER;
  for (int j = 0; j < L_SEQ; ++j) {
    const float* row = xd + (size_t)j * 64;  // [dts(32) | B(16) | C(16)]
    float draw = bias;
#pragma unroll
    for (int r = 0; r < DT_RANK; ++r) draw += W[r] * row[r];
    float delta = (draw > 20.f) ? draw : log1pf(__expf(draw));
    int p = perm_idx(k, j);
    float u = xu[(size_t)p * D_INNER + d];
    float du = delta * u;
    float y = Dd * u;
#pragma unroll
    for (int n = 0; n < D_STATE; ++n) {
      float bn = row[32 + n];
      float cn = row[48 + n];
      h[n] = __expf(delta * A[n]) * h[n] + du * bn;
      y += h[n] * cn;
    }
    yo[(size_t)p * D_INNER + d] = y;
  }
}

// ---------------- K6: merge 4 directions + LayerNorm + gate --------------
__global__ __launch_bounds__(256) void k_merge_ln_gate(
    const float* __restrict__ ypart, const float* __restrict__ gamma,
    const float* __restrict__ beta, float* __restrict__ zbuf) {
  __shared__ float ssum[256], ssq[256];
  int t = blockIdx.x;
  int b = t >> 11;
  int l = t & (L_SEQ - 1);
  int tid = threadIdx.x;
  float yv[4];
  float s = 0.f, sq = 0.f;
#pragma unroll
  for (int q = 0; q < 4; ++q) {
    int d = tid + q * 256;
    float acc = 0.f;
#pragma unroll
    for (int k = 0; k < 4; ++k)
      acc += ypart[((size_t)(k * B_SZ + b) * L_SEQ + l) * D_INNER + d];
    yv[q] = acc; s += acc; sq += acc * acc;
  }
  ssum[tid] = s; ssq[tid] = sq;
  __syncthreads();
  for (int off = 128; off > 0; off >>= 1) {
    if (tid < off) { ssum[tid] += ssum[tid + off]; ssq[tid] += ssq[tid + off]; }
    __syncthreads();
  }
  float mean = ssum[0] * (1.f / D_INNER);
  float var = ssq[0] * (1.f / D_INNER) - mean * mean;
  float rstd = rsqrtf(var + 1e-5f);
#pragma unroll
  for (int q = 0; q < 4; ++q) {
    int d = tid + q * 256;
    float o = (yv[q] - mean) * rstd * gamma[d] + beta[d];
    size_t zi = (size_t)t * D_INNER + d;
    zbuf[zi] = zbuf[zi] * o;   // silu(z) * LN(y), in place over z buffer
  }
}

// ---------------- K7: out_proj GEMM (8192 x 512 x 1024) ------------------
__global__ __launch_bounds__(128) void k_out_proj(
    const float* __restrict__ yg, const float* __restrict__ wo,
    float* __restrict__ out) {
  int wave = threadIdx.x >> 5, lane = threadIdx.x & 31;
  int tile = blockIdx.x * 4 + wave;   // 4096 waves total
  int nt = tile & 15;                 // 512/32 col super-tiles
  int mt = tile >> 4;                 // 8192/32 row super-tiles
  int m0 = mt * 32, n0 = nt * 32;
  int lmod = lane & 15, lhalf = lane >> 4;
  const float* a0 = yg + (size_t)(m0 + lmod) * D_INNER + 2 * lhalf;
  const float* a1 = yg + (size_t)(m0 + 16 + lmod) * D_INNER + 2 * lhalf;
  const float* b0 = wo + (size_t)(n0 + lmod) * D_INNER + 2 * lhalf;
  const float* b1 = wo + (size_t)(n0 + 16 + lmod) * D_INNER + 2 * lhalf;
  v8f acc00 = {0.f,0.f,0.f,0.f,0.f,0.f,0.f,0.f};
  v8f acc01 = acc00, acc10 = acc00, acc11 = acc00;
#pragma unroll 2
  for (int kk = 0; kk < D_INNER; kk += 4) {
    v2f va0 = *(const v2f*)(a0 + kk);
    v2f va1 = *(const v2f*)(a1 + kk);
    v2f vb0 = *(const v2f*)(b0 + kk);
    v2f vb1 = *(const v2f*)(b1 + kk);
    acc00 = wmma_f32_k4(va0, vb0, acc00);
    acc01 = wmma_f32_k4(va0, vb1, acc01);
    acc10 = wmma_f32_k4(va1, vb0, acc10);
    acc11 = wmma_f32_k4(va1, vb1, acc11);
  }
  v8f accs[4] = {acc00, acc01, acc10, acc11};
#pragma unroll
  for (int s = 0; s < 4; ++s) {
    int mb = s >> 1, nb = s & 1;
#pragma unroll
    for (int r = 0; r < 8; ++r) {
      int row = m0 + mb * 16 + 8 * lhalf + r;
      out[(size_t)row * D_MODEL + n0 + nb * 16 + lmod] = accs[s][r];
    }
  }
}

extern "C" void kernel_launch(void* const* d_in, const int* in_sizes, int n_in,
                              void* d_out, int out_size, void* d_ws,
                              size_t ws_size, hipStream_t stream) {
  (void)in_sizes; (void)n_in; (void)out_size; (void)ws_size;
  const float* x    = (const float*)d_in[0];
  const float* inw  = (const float*)d_in[1];
  const float* cw   = (const float*)d_in[2];
  const float* cb   = (const float*)d_in[3];
  const float* wx   = (const float*)d_in[4];
  const float* dtw  = (const float*)d_in[5];
  const float* dtb  = (const float*)d_in[6];
  const float* alog = (const float*)d_in[7];
  const float* dsv  = (const float*)d_in[8];
  const float* gam  = (const float*)d_in[9];
  const float* bet  = (const float*)d_in[10];
  const float* wo   = (const float*)d_in[11];
  float* out = (float*)d_out;
  float* ws = (float*)d_ws;
  // workspace layout (floats):
  float* z_silu = ws;                            //  8M  (32 MB)
  float* xcl    = ws + (size_t)8  * 1024 * 1024; //  8M  (32 MB)
  float* x_dbl  = ws + (size_t)16 * 1024 * 1024; //  2M  ( 8 MB)
  float* big    = ws + (size_t)18 * 1024 * 1024; // 32M  (128 MB)
  float* x_ssm = big;   // live K1..K2 (first 32 MB of big)
  float* ypart = big;   // live K5..K6 (full 128 MB, after x_ssm is dead)

  k_in_proj      <<< 4096, 128, 0, stream>>>(x, inw, x_ssm, z_silu);
  k_conv         <<<32768, 256, 0, stream>>>(x_ssm, cw, cb, xcl);
  k_x_proj       <<<  512, 128, 0, stream>>>(xcl, wx, x_dbl);
  k_scan         <<<   64, 256, 0, stream>>>(xcl, x_dbl, dtw, dtb, alog, dsv,
                                             ypart);
  k_merge_ln_gate<<< 8192, 256, 0, stream>>>(ypart, gam, bet, z_silu);
  k_out_proj     <<< 1024, 128, 0, stream>>>(z_silu, wo, out);
}